// VectorQuantizer_17935783428445
// MI455X (gfx1250) — compile-verified
//
#include <hip/hip_runtime.h>
#include <hip/hip_bf16.h>
#include <math.h>

typedef __attribute__((ext_vector_type(2))) float v2f;
typedef __attribute__((ext_vector_type(8))) float v8f;

#define EMB_DIM    256
#define KCODES     1024
#define N_ROWS     131072            // 32 * 64 * 64
#define HW         4096              // 64*64
#define CHW        1048576           // 256*4096
#define A_STRIDE   260               // 260 mod 64 == 4 -> conflict-free 16x4 A reads
#define D_STRIDE   1028              // rows r and r+8 land 32 banks apart
#define OUT_N      8388608           // 32*256*64*64

// dynamic LDS partition (in dwords):
//   A tile      : 16*A_STRIDE = 4160
//   a2 (||z||^2): 16
//   cbn cache   : 1024
//   d tile      : 16*D_STRIDE = 16448
//   cand values : 8 waves * 256 = 2048
//   cand index  : 2048 (ints)
//   minidx      : 16 (ints)
#define SMEM_DWORDS (4160 + 16 + 1024 + 16448 + 2048 + 2048 + 16)
#define SMEM_BYTES  (SMEM_DWORDS * 4)

__global__ void vq_init(double* sum_d, double* sum_sq, int* hist) {
    int t = threadIdx.x;
    if (t < KCODES) hist[t] = 0;
    if (t == 0) { *sum_d = 0.0; *sum_sq = 0.0; }
}

__global__ void vq_cbnorm(const float* __restrict__ cb, float* __restrict__ cbn) {
    int k = blockIdx.x * 256 + threadIdx.x;        // 4 blocks x 256 = 1024 codes
    const float4* p = (const float4*)(cb + (size_t)k * EMB_DIM);
    float s = 0.f;
    #pragma unroll 8
    for (int i = 0; i < EMB_DIM / 4; ++i) {
        float4 v = p[i];
        s += v.x * v.x + v.y * v.y + v.z * v.z + v.w * v.w;
    }
    cbn[k] = s;
}

__global__ void vq_main(const float* __restrict__ z,
                        const float* __restrict__ cb,
                        const float* __restrict__ cbn_g,
                        float* __restrict__ out,
                        float* __restrict__ minidx_out,
                        float* __restrict__ topk_out,
                        double* __restrict__ sum_d,
                        double* __restrict__ sum_sq,
                        int* __restrict__ hist) {
    extern __shared__ char smem_raw[];
    float* As   = (float*)smem_raw;                 // [16][A_STRIDE]
    float* a2   = As + 16 * A_STRIDE;               // [16]
    float* cbn  = a2 + 16;                          // [1024]
    float* dl   = cbn + KCODES;                     // [16][D_STRIDE]
    float* candv = dl + 16 * D_STRIDE;              // [8][256]
    int*   candi = (int*)(candv + 8 * 256);         // [8][256]
    int*   minidx = candi + 8 * 256;                // [16]

    const int t    = threadIdx.x;
    const int lane = t & 31;
    const int wave = t >> 5;
    const int half = lane >> 4;                     // k-half select for WMMA layout
    const int l16  = lane & 15;

    const int n0 = blockIdx.x * 16;                 // first flat (b,h,w) row
    const int b  = n0 >> 12;
    const int s0 = n0 & (HW - 1);
    const size_t zbase = (size_t)b * CHW;

    // ---- stage A tile (16 rows x 256 ch), coalesced along spatial dim ----
    {
        int srow = t & 15;
        int c0   = t >> 4;
        #pragma unroll
        for (int i = 0; i < 16; ++i) {
            int c = c0 + 16 * i;
            As[srow * A_STRIDE + c] = z[zbase + (size_t)c * HW + s0 + srow];
        }
    }
    for (int i = t; i < KCODES; i += 256) cbn[i] = cbn_g[i];
    __syncthreads();

    if (t < 16) {                                   // ||z_row||^2
        float s = 0.f;
        for (int c = 0; c < EMB_DIM; ++c) { float v = As[t * A_STRIDE + c]; s += v * v; }
        a2[t] = s;
    }
    __syncthreads();

    // ---- fp32 WMMA GEMM: each wave owns 128 codes (8 col-tiles of 16) ----
    float dsum = 0.f;
    for (int ct = 0; ct < 8; ++ct) {
        const int col = wave * 128 + ct * 16 + l16;
        v8f acc = {};
        const float* bp = cb + (size_t)col * EMB_DIM + 2 * half;   // B: 4x16, K from codebook row
        const float* ap = As + l16 * A_STRIDE + 2 * half;          // A: 16x4 from LDS
        #pragma unroll 8
        for (int kb = 0; kb < EMB_DIM / 4; ++kb) {
            v2f a = *(const v2f*)(ap + kb * 4);
            v2f bb = *(const v2f*)(bp + kb * 4);
            acc = __builtin_amdgcn_wmma_f32_16x16x4_f32(
                false, a, false, bb, (short)0, acc, false, false);
        }
        const float b2c = cbn[col];
        #pragma unroll
        for (int r = 0; r < 8; ++r) {
            int row = r + 8 * half;                 // C/D layout: VGPR r -> row r (+8 for hi lanes)
            float dv = a2[row] + b2c - 2.0f * acc[r];
            dl[row * D_STRIDE + col] = dv;
            dsum += dv;
        }
    }
    __syncthreads();

    // ---- per-row top-8 (lexicographic (d, idx), matches argmin/top_k ties) ----
    for (int rr = 0; rr < 2; ++rr) {
        const int row = wave * 2 + rr;
        float bv[8]; int bi[8];
        #pragma unroll
        for (int j = 0; j < 8; ++j) { bv[j] = 3.4e38f; bi[j] = 0x7fffffff; }
        for (int c = lane; c < KCODES; c += 32) {
            float v = dl[row * D_STRIDE + c];
            if (v < bv[7] || (v == bv[7] && c < bi[7])) {
                int j = 7;
                while (j > 0 && (v < bv[j - 1] || (v == bv[j - 1] && c < bi[j - 1]))) {
                    bv[j] = bv[j - 1]; bi[j] = bi[j - 1]; --j;
                }
                bv[j] = v; bi[j] = c;
            }
        }
        float* cv = candv + wave * 256;
        int*   ci = candi + wave * 256;
        #pragma unroll
        for (int j = 0; j < 8; ++j) { cv[lane * 8 + j] = bv[j]; ci[lane * 8 + j] = bi[j]; }
        __syncthreads();
        if (lane == 0) {
            float tv[8]; int ti[8];
            #pragma unroll
            for (int j = 0; j < 8; ++j) { tv[j] = 3.4e38f; ti[j] = 0x7fffffff; }
            for (int m = 0; m < 256; ++m) {
                float v = cv[m]; int c = ci[m];
                if (v < tv[7] || (v == tv[7] && c < ti[7])) {
                    int j = 7;
                    while (j > 0 && (v < tv[j - 1] || (v == tv[j - 1] && c < ti[j - 1]))) {
                        tv[j] = tv[j - 1]; ti[j] = ti[j - 1]; --j;
                    }
                    tv[j] = v; ti[j] = c;
                }
            }
            const int n = n0 + row;
            minidx_out[n] = (float)ti[0];
            #pragma unroll
            for (int j = 0; j < 8; ++j) topk_out[(size_t)n * 8 + j] = (float)ti[j];
            minidx[row] = ti[0];
            atomicAdd(&hist[ti[0]], 1);
        }
        __syncthreads();
    }

    // ---- z_q gather, NCHW output (coalesced along spatial), MSE partial ----
    float lsum = 0.f;
    {
        int srow = t & 15;
        int c0   = t >> 4;
        int idx  = minidx[srow];
        const float* qrow = cb + (size_t)idx * EMB_DIM;
        #pragma unroll
        for (int i = 0; i < 16; ++i) {
            int c = c0 + 16 * i;
            float q  = qrow[c];
            float zv = As[srow * A_STRIDE + c];
            float df = q - zv;
            lsum += df * df;
            out[zbase + (size_t)c * HW + s0 + srow] = q;   // straight-through fwd == z_q
        }
    }

    // ---- block reductions -> global double accumulators ----
    __syncthreads();
    float* red = candv;                              // reuse scratch
    red[t] = dsum; __syncthreads();
    for (int ofs = 128; ofs > 0; ofs >>= 1) {
        if (t < ofs) red[t] += red[t + ofs];
        __syncthreads();
    }
    if (t == 0) atomicAdd(sum_d, (double)red[0]);
    __syncthreads();
    red[t] = lsum; __syncthreads();
    for (int ofs = 128; ofs > 0; ofs >>= 1) {
        if (t < ofs) red[t] += red[t + ofs];
        __syncthreads();
    }
    if (t == 0) atomicAdd(sum_sq, (double)red[0]);
}

__global__ void vq_finalize(const double* __restrict__ sum_d,
                            const double* __restrict__ sum_sq,
                            const int* __restrict__ hist,
                            float* __restrict__ scalars) {
    if (threadIdx.x == 0) {
        double ss = *sum_sq;
        double sd = *sum_d;
        scalars[0] = (float)(1.25 * ss / ((double)N_ROWS * (double)EMB_DIM));  // loss
        double ent = 0.0;
        for (int k = 0; k < KCODES; ++k) {
            double e = (double)hist[k] / (double)N_ROWS;
            ent += e * log(e + 1e-10);
        }
        scalars[1] = (float)exp(-ent);                                          // perplexity
        scalars[2] = (float)(sd / ((double)N_ROWS * (double)KCODES));           // mean_distance
    }
}

extern "C" void kernel_launch(void* const* d_in, const int* in_sizes, int n_in,
                              void* d_out, int out_size, void* d_ws, size_t ws_size,
                              hipStream_t stream) {
    const float* z  = (const float*)d_in[0];
    const float* cb = (const float*)d_in[1];
    float* out = (float*)d_out;

    char* ws = (char*)d_ws;
    double* sum_d  = (double*)ws;
    double* sum_sq = (double*)(ws + 8);
    int*    hist   = (int*)(ws + 16);
    float*  cbn    = (float*)(ws + 16 + KCODES * 4);

    float* scalars    = out + OUT_N;          // loss, perplexity, mean_distance
    float* minidx_out = out + OUT_N + 3;      // [N_ROWS]
    float* topk_out   = minidx_out + N_ROWS;  // [N_ROWS * 8]

    vq_init<<<1, 1024, 0, stream>>>(sum_d, sum_sq, hist);
    vq_cbnorm<<<4, 256, 0, stream>>>(cb, cbn);
    vq_main<<<N_ROWS / 16, 256, SMEM_BYTES, stream>>>(
        z, cb, cbn, out, minidx_out, topk_out, sum_d, sum_sq, hist);
    vq_finalize<<<1, 32, 0, stream>>>(sum_d, sum_sq, hist, scalars);
}